// PostProcessingModule_11879879543434
// MI455X (gfx1250) — compile-verified
//
#include <hip/hip_runtime.h>
#include <hip/hip_bf16.h>

// ---------------- problem constants (match reference) ----------------
constexpr int  kB        = 4;
constexpr int  kN        = 4096;
constexpr int  kC        = 80;
constexpr int  kND       = 5 + kC;      // 85
constexpr int  kMaxDet   = 300;
constexpr int  kTile     = 128;         // rows per tile == rows per wave (32 lanes * 4)
constexpr int  kTiles    = kN / kTile;  // 32 == number of waves in the block
constexpr float kClsOff  = 4.0f;
constexpr float kEps     = 1e-7f;

#if defined(__has_builtin)
#if __has_builtin(__builtin_amdgcn_global_load_async_to_lds_b128)
#define HAS_ASYNC_LDS 1
#endif
#endif

#ifdef HAS_ASYNC_LDS
typedef int v4i_t __attribute__((vector_size(4 * sizeof(int))));
typedef __attribute__((address_space(1))) v4i_t g_v4i;   // global
typedef __attribute__((address_space(3))) v4i_t l_v4i;   // LDS
#endif

// =====================================================================
// Kernel 1: decode boxes, per-box class score max/argmax, threshold.
// =====================================================================
__global__ void yolo_decode_kernel(const float* __restrict__ yolo,
                                   const float* __restrict__ score_thr_p,
                                   float* __restrict__ boxes,   // [B*N*4]
                                   float* __restrict__ score,   // [B*N]
                                   float* __restrict__ clsf)    // [B*N]
{
    int gid = blockIdx.x * blockDim.x + threadIdx.x;
    if (gid >= kB * kN) return;

    const float* p = yolo + (size_t)gid * kND;
    float cx = p[0], cy = p[1], w = p[2], h = p[3], obj = p[4];

    // argmax over obj*cls, first-max wins (jnp.argmax semantics)
    float best = -1.0f; int bi = 0;
    for (int k = 0; k < kC; ++k) {
        float v = obj * p[5 + k];
        if (v > best) { best = v; bi = k; }
    }
    float sthr = *score_thr_p;
    float s = (best > sthr) ? best : 0.0f;   // where(valid, score, 0)

    float* bo = boxes + (size_t)gid * 4;
    bo[0] = cx - 0.5f * w;
    bo[1] = cy - 0.5f * h;
    bo[2] = cx + 0.5f * w;
    bo[3] = cy + 0.5f * h;
    score[gid] = s;
    clsf[gid]  = (float)bi;
}

// select register-array element by uniform 2-bit slot (keeps data in VGPRs)
#define SEL4(arr, sl) ((sl) == 0 ? arr[0] : (sl) == 1 ? arr[1] : (sl) == 2 ? arr[2] : arr[3])

// =====================================================================
// Kernel 2: per-batch sort + tiled greedy NMS + compaction.
// One 1024-thread (32-wave) workgroup per batch; 4 rows per thread.
// Tile t (sorted rows [128t,128t+128)) is owned by wave t: the strictly
// sequential greedy scan runs on that single wave via __shfl (no
// barriers); surviving rows are then applied to all later rows by the
// whole block. 2 barriers per tile instead of 2 per row.
// =====================================================================
__global__ __launch_bounds__(1024)
void yolo_nms_kernel(const float* __restrict__ boxes,  // [B*N*4]
                     const float* __restrict__ score,  // [B*N]
                     const float* __restrict__ clsf,   // [B*N]
                     const float* __restrict__ iou_thr_p,
                     float* __restrict__ out)          // [B*300*6]
{
    const int b    = blockIdx.x;
    const int tid  = threadIdx.x;
    const int lane = tid & 31;
    const int wv   = tid >> 5;

    __shared__ unsigned long long s_key[kN];   // 32 KB sort keys
    __shared__ float s_sc[kN];                 // 16 KB async-staged scores
    __shared__ float s_tx1[kTile], s_ty1[kTile], s_tx2[kTile], s_ty2[kTile], s_ta[kTile];
    __shared__ int   s_tkeep[kTile];
    __shared__ int   s_anyk;
    __shared__ int   s_cnt[1024];

    const float ithr = *iou_thr_p;

    // -------- zero this batch's output region early --------
    float* outb = out + (size_t)b * kMaxDet * 6;
    for (int t = tid; t < kMaxDet * 6; t += 1024) outb[t] = 0.0f;
    __threadfence_block();

    // -------- Phase A: stage scores to LDS (async global->LDS) --------
    {
        const float* g = score + (size_t)b * kN + tid * 4;   // 16B aligned
        float*       l = &s_sc[tid * 4];
#ifdef HAS_ASYNC_LDS
        __builtin_amdgcn_global_load_async_to_lds_b128(
            (g_v4i*)g, (l_v4i*)l, 0, 0);
        asm volatile("s_wait_asynccnt 0" ::: "memory");
#else
        l[0] = g[0]; l[1] = g[1]; l[2] = g[2]; l[3] = g[3];
#endif
    }
    __syncthreads();

    // -------- Phase B: build keys, bitonic sort descending --------
    // key = (score_bits << 32) | (0xFFFFFFFF - idx); scores >= 0 so bit
    // pattern is order-preserving; descending == stable argsort(-score).
    #pragma unroll
    for (int u = 0; u < 4; ++u) {
        int p = tid * 4 + u;
        unsigned sb = __float_as_uint(s_sc[p]);
        s_key[p] = ((unsigned long long)sb << 32) |
                   (unsigned long long)(0xFFFFFFFFu - (unsigned)p);
    }

    for (unsigned k = 2; k <= (unsigned)kN; k <<= 1) {
        for (unsigned j = k >> 1; j > 0; j >>= 1) {
            __syncthreads();
            for (unsigned ii = tid; ii < (unsigned)kN; ii += 1024) {
                unsigned l = ii ^ j;
                if (l > ii) {
                    unsigned long long a = s_key[ii];
                    unsigned long long c = s_key[l];
                    bool swp = ((ii & k) == 0) ? (a < c) : (a > c);
                    if (swp) { s_key[ii] = c; s_key[l] = a; }
                }
            }
        }
    }
    __syncthreads();

    // -------- Phase C: gather my 4 sorted rows into registers --------
    float rbx0[4], rbx1[4], rbx2[4], rbx3[4];   // raw boxes (output)
    float ox0[4],  ox1[4],  ox2[4],  ox3[4];    // class-offset boxes (IoU)
    float areas[4], scs[4], cl4[4];
    bool  kps[4];

    #pragma unroll
    for (int u = 0; u < 4; ++u) {
        int p = tid * 4 + u;
        unsigned long long key = s_key[p];
        float    sc  = __uint_as_float((unsigned)(key >> 32));
        unsigned idx = 0xFFFFFFFFu - (unsigned)(key & 0xFFFFFFFFull);
        const float* bo = boxes + ((size_t)b * kN + idx) * 4;
        float c  = clsf[(size_t)b * kN + idx];
        float x1 = bo[0], y1 = bo[1], x2 = bo[2], y2 = bo[3];
        float off = c * kClsOff;
        rbx0[u] = x1; rbx1[u] = y1; rbx2[u] = x2; rbx3[u] = y2;
        ox0[u] = x1 + off; ox1[u] = y1 + off;
        ox2[u] = x2 + off; ox3[u] = y2 + off;
        areas[u] = (ox2[u] - ox0[u]) * (ox3[u] - ox1[u]);   // reference order
        scs[u] = sc; cl4[u] = c;
        kps[u] = (sc > 0.0f);       // valid == kept at init
    }

    // -------- Phase D: tiled greedy suppression --------
    for (int t = 0; t < kTiles; ++t) {
        __syncthreads();   // previous tile's apply done; safe to reuse s_t*

        if (wv == t) {
            // publish my tile rows' coords (coords never change)
            #pragma unroll
            for (int u = 0; u < 4; ++u) {
                int r = lane * 4 + u;
                s_tx1[r] = ox0[u]; s_ty1[r] = ox1[u];
                s_tx2[r] = ox2[u]; s_ty2[r] = ox3[u];
                s_ta[r]  = areas[u];
            }
            // single-wave sequential greedy scan over 128 rows via shfl
            for (int i = 0; i < kTile; ++i) {
                int src = i >> 2, sl = i & 3;
                float bx1 = __shfl(SEL4(ox0, sl),   src, 32);
                float by1 = __shfl(SEL4(ox1, sl),   src, 32);
                float bx2 = __shfl(SEL4(ox2, sl),   src, 32);
                float by2 = __shfl(SEL4(ox3, sl),   src, 32);
                float ba  = __shfl(SEL4(areas, sl), src, 32);
                int km = (kps[0] ? 1 : 0) | (kps[1] ? 2 : 0) |
                         (kps[2] ? 4 : 0) | (kps[3] ? 8 : 0);
                int ki = (__shfl(km, src, 32) >> sl) & 1;
                if (ki) {
                    #pragma unroll
                    for (int u = 0; u < 4; ++u) {
                        int r = lane * 4 + u;       // row-in-tile
                        if (r > i && kps[u]) {
                            float ltx = fmaxf(bx1, ox0[u]);
                            float lty = fmaxf(by1, ox1[u]);
                            float rbx = fminf(bx2, ox2[u]);
                            float rby = fminf(by2, ox3[u]);
                            float iw = fmaxf(rbx - ltx, 0.0f);
                            float ih = fmaxf(rby - lty, 0.0f);
                            float inter = iw * ih;
                            float iou = inter / (ba + areas[u] - inter + kEps);
                            if (iou > ithr) kps[u] = false;
                        }
                    }
                }
            }
            // publish final keep flags for this tile
            int km = (kps[0] ? 1 : 0) | (kps[1] ? 2 : 0) |
                     (kps[2] ? 4 : 0) | (kps[3] ? 8 : 0);
            #pragma unroll
            for (int u = 0; u < 4; ++u) s_tkeep[lane * 4 + u] = (km >> u) & 1;
            unsigned long long bal = __ballot(km != 0);
            if (lane == 0) s_anyk = (bal != 0ull) ? 1 : 0;
        }
        __syncthreads();   // tile coords + keep flags visible

        // apply surviving tile rows to all later rows (rows >= (t+1)*128,
        // i.e. threads tid >= (t+1)*32); earlier rows are already final.
        if (s_anyk && tid >= (t + 1) * 32) {
            for (int i = 0; i < kTile; ++i) {
                if (!s_tkeep[i]) continue;
                float bx1 = s_tx1[i], by1 = s_ty1[i];
                float bx2 = s_tx2[i], by2 = s_ty2[i];
                float ba  = s_ta[i];
                #pragma unroll
                for (int u = 0; u < 4; ++u) {
                    if (kps[u]) {
                        float ltx = fmaxf(bx1, ox0[u]);
                        float lty = fmaxf(by1, ox1[u]);
                        float rbx = fminf(bx2, ox2[u]);
                        float rby = fminf(by2, ox3[u]);
                        float iw = fmaxf(rbx - ltx, 0.0f);
                        float ih = fmaxf(rby - lty, 0.0f);
                        float inter = iw * ih;
                        float iou = inter / (ba + areas[u] - inter + kEps);
                        if (iou > ithr) kps[u] = false;
                    }
                }
            }
        }
    }
    __syncthreads();

    // -------- Phase E: compaction (block prefix scan) --------
    int c = (kps[0] ? 1 : 0) + (kps[1] ? 1 : 0) +
            (kps[2] ? 1 : 0) + (kps[3] ? 1 : 0);
    s_cnt[tid] = c;
    __syncthreads();
    for (int d = 1; d < 1024; d <<= 1) {
        int add = (tid >= d) ? s_cnt[tid - d] : 0;
        __syncthreads();
        s_cnt[tid] += add;
        __syncthreads();
    }
    int rank = s_cnt[tid] - c;   // exclusive prefix

    #pragma unroll
    for (int u = 0; u < 4; ++u) {
        if (kps[u]) {
            if (rank < kMaxDet) {
                float* o = outb + (size_t)rank * 6;
                o[0] = rbx0[u]; o[1] = rbx1[u];
                o[2] = rbx2[u]; o[3] = rbx3[u];
                o[4] = scs[u];  o[5] = cl4[u];
            }
            ++rank;
        }
    }
}

// =====================================================================
extern "C" void kernel_launch(void* const* d_in, const int* in_sizes, int n_in,
                              void* d_out, int out_size, void* d_ws, size_t ws_size,
                              hipStream_t stream) {
    const float* yolo      = (const float*)d_in[0];
    const float* iou_thr   = (const float*)d_in[1];
    const float* score_thr = (const float*)d_in[2];
    float* out = (float*)d_out;

    float* wsf   = (float*)d_ws;
    float* boxes = wsf;                          // B*N*4 floats
    float* score = wsf + (size_t)kB * kN * 4;    // B*N floats
    float* clsf  = wsf + (size_t)kB * kN * 5;    // B*N floats

    int total = kB * kN;
    yolo_decode_kernel<<<(total + 255) / 256, 256, 0, stream>>>(
        yolo, score_thr, boxes, score, clsf);

    yolo_nms_kernel<<<kB, 1024, 0, stream>>>(
        boxes, score, clsf, iou_thr, out);
}